// KVCache_3100966387968
// MI455X (gfx1250) — compile-verified
//
#include <hip/hip_runtime.h>
#include <hip/hip_bf16.h>
#include <stdint.h>

// Problem constants (match reference)
constexpr int BS = 16, G = 8, S = 4096, H = 128, T = 2048;
constexpr int S_LOG2 = 12;          // S == 4096
constexpr int ROWS_PER_BLOCK = 8;   // 256 threads = 8 wave32, one row per wave

// ---------------------------------------------------------------------------
// inv[s] = index of the LAST t with input_pos[t] == s, else -1  (index_copy
// "last write wins" semantics via atomicMax).
// ---------------------------------------------------------------------------
__global__ void inv_init(int* __restrict__ inv) {
    inv[blockIdx.x * blockDim.x + threadIdx.x] = -1;
}

__global__ void inv_scatter(const int* __restrict__ pos, int* __restrict__ inv, int n) {
    int t = blockIdx.x * blockDim.x + threadIdx.x;
    if (t < n) {
        int p = pos[t];
        if (p >= 0 && p < S) atomicMax(&inv[p], t);
    }
}

// ---------------------------------------------------------------------------
// Async DMA of one 512B row pair (k,v) through LDS: CDNA5
// global_load_async_to_lds_b128 / global_store_async_from_lds_b128.
// Data never touches VGPRs; tracked with ASYNCcnt. Non-temporal hints keep
// this 1 GB one-shot stream from thrashing the 192 MB L2.
// Each lane moves 16B; a wave32 moves one full 128-float row per op.
// ---------------------------------------------------------------------------
__device__ __forceinline__ void async_row_copy(const float* gk, const float* gv,
                                               float* dk, float* dv,
                                               unsigned ldsK, unsigned ldsV) {
    asm volatile("global_load_async_to_lds_b128 %0, %1, off th:TH_LOAD_NT"
                 :: "v"(ldsK), "v"(gk) : "memory");
    asm volatile("global_load_async_to_lds_b128 %0, %1, off th:TH_LOAD_NT"
                 :: "v"(ldsV), "v"(gv) : "memory");
    asm volatile("s_wait_asynccnt 0" ::: "memory");
    asm volatile("global_store_async_from_lds_b128 %0, %1, off th:TH_STORE_NT"
                 :: "v"(dk), "v"(ldsK) : "memory");
    asm volatile("global_store_async_from_lds_b128 %0, %1, off th:TH_STORE_NT"
                 :: "v"(dv), "v"(ldsV) : "memory");
    // implicit S_WAIT_IDLE at S_ENDPGM drains outstanding async stores
}

// ---------------------------------------------------------------------------
// Main pass: one wave per output row s of one (b,g). Source is either the new
// k/v row (inv[s] >= 0) or the cache row. Branch is wave-uniform so EXEC stays
// all-ones (required for the async ops). Total traffic: 1.0 GB -> ~44us @ 23.3TB/s.
// ---------------------------------------------------------------------------
__global__ __launch_bounds__(256) void kv_gather_copy(
    const float* __restrict__ k,  const float* __restrict__ v,
    const float* __restrict__ ck, const float* __restrict__ cv,
    const int* __restrict__ inv,
    float* __restrict__ ok, float* __restrict__ ov)
{
    __shared__ float lds[2 * 256 * 4];   // 8KB: [0..1023]=k stage, [1024..2047]=v stage
    const unsigned tid  = threadIdx.x;
    const unsigned lane = tid & 31u;
    const unsigned wave = tid >> 5;
    const unsigned row  = blockIdx.x * ROWS_PER_BLOCK + wave;  // < BS*G*S
    const unsigned s    = row & (S - 1);
    const unsigned bg   = row >> S_LOG2;

    const int tv = inv[s];               // wave-uniform
    const float* srck;
    const float* srcv;
    if (tv >= 0) {
        const size_t off = ((size_t)bg * T + (unsigned)tv) * H;
        srck = k + off;  srcv = v + off;
    } else {
        const size_t off = ((size_t)bg * S + s) * H;
        srck = ck + off; srcv = cv + off;
    }
    const size_t doff = ((size_t)bg * S + s) * H + lane * 4u;

    // Flat shared addr: high 32 bits = LDS aperture, low 32 bits = LDS byte offset.
    const unsigned ldsK = (unsigned)(uintptr_t)&lds[tid * 4u];
    const unsigned ldsV = (unsigned)(uintptr_t)&lds[1024u + tid * 4u];

    async_row_copy(srck + lane * 4u, srcv + lane * 4u, ok + doff, ov + doff, ldsK, ldsV);
}

// ---------------------------------------------------------------------------
// Fallback (only if d_ws can't hold inv[]): full cache copy + scatter (1.5 GB).
// ---------------------------------------------------------------------------
__global__ __launch_bounds__(256) void kv_copy_cache(
    const float* __restrict__ ck, const float* __restrict__ cv,
    float* __restrict__ ok, float* __restrict__ ov)
{
    __shared__ float lds[2 * 256 * 4];
    const unsigned tid  = threadIdx.x;
    const unsigned lane = tid & 31u;
    const unsigned row  = blockIdx.x * ROWS_PER_BLOCK + (tid >> 5);
    const size_t   off  = (size_t)row * H + lane * 4u;
    const unsigned ldsK = (unsigned)(uintptr_t)&lds[tid * 4u];
    const unsigned ldsV = (unsigned)(uintptr_t)&lds[1024u + tid * 4u];
    async_row_copy(ck + off, cv + off, ok + off, ov + off, ldsK, ldsV);
}

__global__ __launch_bounds__(256) void kv_scatter(
    const int* __restrict__ pos,
    const float* __restrict__ k, const float* __restrict__ v,
    float* __restrict__ ok, float* __restrict__ ov)
{
    __shared__ float lds[2 * 256 * 4];
    const unsigned tid  = threadIdx.x;
    const unsigned lane = tid & 31u;
    const unsigned row  = blockIdx.x * ROWS_PER_BLOCK + (tid >> 5); // < BS*G*T
    const unsigned t    = row % T;
    const unsigned bg   = row / T;
    const int s = pos[t];                                           // wave-uniform
    const size_t soff = ((size_t)bg * T + t) * H + lane * 4u;
    const size_t doff = ((size_t)bg * S + (unsigned)s) * H + lane * 4u;
    const unsigned ldsK = (unsigned)(uintptr_t)&lds[tid * 4u];
    const unsigned ldsV = (unsigned)(uintptr_t)&lds[1024u + tid * 4u];
    async_row_copy(k + soff, v + soff, ok + doff, ov + doff, ldsK, ldsV);
}

extern "C" void kernel_launch(void* const* d_in, const int* in_sizes, int n_in,
                              void* d_out, int out_size, void* d_ws, size_t ws_size,
                              hipStream_t stream) {
    const int*   pos = (const int*)  d_in[0];   // input_pos (int32)
    const float* k   = (const float*)d_in[1];
    const float* v   = (const float*)d_in[2];
    const float* ck  = (const float*)d_in[3];
    const float* cv  = (const float*)d_in[4];
    float* ok = (float*)d_out;
    float* ov = ok + (size_t)BS * G * S * H;

    const int nT = in_sizes[0];                 // == T

    if (ws_size >= (size_t)S * sizeof(int)) {
        int* inv = (int*)d_ws;
        inv_init   <<<S / 256, 256, 0, stream>>>(inv);
        inv_scatter<<<(nT + 255) / 256, 256, 0, stream>>>(pos, inv, nT);
        kv_gather_copy<<<(BS * G * S) / ROWS_PER_BLOCK, 256, 0, stream>>>(
            k, v, ck, cv, inv, ok, ov);
    } else {
        kv_copy_cache<<<(BS * G * S) / ROWS_PER_BLOCK, 256, 0, stream>>>(ck, cv, ok, ov);
        kv_scatter   <<<(BS * G * nT) / ROWS_PER_BLOCK, 256, 0, stream>>>(pos, k, v, ok, ov);
    }
}